// MLPTokenizer_90477781057544
// MI455X (gfx1250) — compile-verified
//
#include <hip/hip_runtime.h>
#include <hip/hip_bf16.h>

typedef float v2f __attribute__((ext_vector_type(2)));
typedef float v8f __attribute__((ext_vector_type(8)));

#define NFEAT 100
#define DTOK  192
#define DHID  96
#define BATCH 4096
#define MTILE 64          // batch rows per workgroup
#define LDS_STRIDE 100    // padded row stride (floats): conflict-free column reads, keeps b64 align

// 0.5*v*(1 + erf(v/sqrt(2))), erf via Abramowitz–Stegun 7.1.26 (|err| < 1.5e-7)
// ~15 VALU ops incl. one v_rcp_f32 and one v_exp_f32 (TRANS pipe) vs ~30 for libm erff.
__device__ __forceinline__ float gelu_exact(float v) {
    float u = v * 0.70710678118654752440f;
    float a = __builtin_fabsf(u);
    float t = __builtin_amdgcn_rcpf(__builtin_fmaf(0.3275911f, a, 1.0f));
    float p = __builtin_fmaf(t, 1.061405429f, -1.453152027f);
    p = __builtin_fmaf(t, p, 1.421413741f);
    p = __builtin_fmaf(t, p, -0.284496736f);
    p = __builtin_fmaf(t, p, 0.254829592f);
    p = p * t;
    float e = __expf(-a * a);                    // v_exp_f32
    float erfa = __builtin_fmaf(-p, e, 1.0f);
    float erfu = __builtin_copysignf(erfa, u);
    return 0.5f * v * (1.0f + erfu);
}

__global__ __launch_bounds__(256) void mlp_tokenizer_wmma(
    const float* __restrict__ x,   // [B, F]
    const float* __restrict__ W1,  // [F, H]
    const float* __restrict__ b1,  // [F, H]
    const float* __restrict__ W2,  // [F, H, D]
    const float* __restrict__ b2,  // [F, D]
    float* __restrict__ out)       // [B, F, D]
{
    __shared__ float hbuf[MTILE * LDS_STRIDE];   // 64 x 96 hidden tile (padded)

    const int f    = blockIdx.y;          // 0..99
    const int b0   = blockIdx.x * MTILE;  // batch block of 64 rows
    const int tid  = threadIdx.x;
    const int wave = tid >> 5;
    const int lane = tid & 31;

    // ---- Phase 1: h[64][96] = gelu(x*W1 + b1) into LDS ----
    // Wave w owns rows m = 8w..8w+7; lane owns cols k = lane, lane+32, lane+64.
    // W1/b1 live in registers (6 loads/thread); x loaded once per row (lane-uniform).
    {
        const int m0 = wave * 8;
        float w1v0 = W1[f * DHID + lane];
        float w1v1 = W1[f * DHID + lane + 32];
        float w1v2 = W1[f * DHID + lane + 64];
        float b1v0 = b1[f * DHID + lane];
        float b1v1 = b1[f * DHID + lane + 32];
        float b1v2 = b1[f * DHID + lane + 64];
        #pragma unroll
        for (int r = 0; r < 8; ++r) {
            float xv = x[(size_t)(b0 + m0 + r) * NFEAT + f];
            float* hr = hbuf + (m0 + r) * LDS_STRIDE + lane;
            hr[0]  = gelu_exact(__builtin_fmaf(xv, w1v0, b1v0));
            hr[32] = gelu_exact(__builtin_fmaf(xv, w1v1, b1v1));
            hr[64] = gelu_exact(__builtin_fmaf(xv, w1v2, b1v2));
        }
    }
    __syncthreads();

    // ---- Phase 2: per-wave 32x48 output tile (2 M-tiles x 3 N-tiles), K=96 in steps of 4 ----
    const int msub  = (wave >> 2) * 32;   // 0 or 32
    const int nbase = (wave & 3) * 48;    // 0,48,96,144
    const int half  = lane >> 4;          // 0: lanes 0-15, 1: lanes 16-31
    const int l16   = lane & 15;
    const int kofs  = half * 2;           // fragment VGPR0 holds K+0 (lo half) / K+2 (hi half)

    v8f acc00 = {}, acc01 = {}, acc02 = {};   // M-tile 0, N-tiles 0..2
    v8f acc10 = {}, acc11 = {}, acc12 = {};   // M-tile 1, N-tiles 0..2

    const float* __restrict__ W2f   = W2 + (size_t)f * (DHID * DTOK);
    const float* __restrict__ hrow0 = hbuf + (msub + l16) * LDS_STRIDE;
    const float* __restrict__ hrow1 = hrow0 + 16 * LDS_STRIDE;

    for (int kk = 0; kk < DHID; kk += 4) {
        // A fragments (16x4 f32): lane = row M=l16, K = kk+kofs .. kk+kofs+1
        v2f a0, a1;
        a0.x = hrow0[kk + kofs];   a0.y = hrow0[kk + kofs + 1];
        a1.x = hrow1[kk + kofs];   a1.y = hrow1[kk + kofs + 1];

        // B fragments (4x16 f32): lane = col N=l16, K rows mirrored like A.
        // Shared across both M-tiles -> amortized 2x. W2 (7.4 MB) stays L2-resident.
        const float* w2k = W2f + (size_t)(kk + kofs) * DTOK + l16;
        v2f bb0, bb1, bb2;
        bb0.x = w2k[nbase +  0];   bb0.y = w2k[DTOK + nbase +  0];
        bb1.x = w2k[nbase + 16];   bb1.y = w2k[DTOK + nbase + 16];
        bb2.x = w2k[nbase + 32];   bb2.y = w2k[DTOK + nbase + 32];

        acc00 = __builtin_amdgcn_wmma_f32_16x16x4_f32(false, a0, false, bb0, (short)0, acc00, false, false);
        acc01 = __builtin_amdgcn_wmma_f32_16x16x4_f32(false, a0, false, bb1, (short)0, acc01, false, false);
        acc02 = __builtin_amdgcn_wmma_f32_16x16x4_f32(false, a0, false, bb2, (short)0, acc02, false, false);
        acc10 = __builtin_amdgcn_wmma_f32_16x16x4_f32(false, a1, false, bb0, (short)0, acc10, false, false);
        acc11 = __builtin_amdgcn_wmma_f32_16x16x4_f32(false, a1, false, bb1, (short)0, acc11, false, false);
        acc12 = __builtin_amdgcn_wmma_f32_16x16x4_f32(false, a1, false, bb2, (short)0, acc12, false, false);
    }

    // ---- bias + store: C VGPR r holds M=r (lanes 0-15) / M=r+8 (lanes 16-31), N=l16 ----
    // Output is 315 MB write-once: non-temporal stores keep L2 for W2/W1/x reuse.
    const float bias0 = b2[f * DTOK + nbase +  0 + l16];
    const float bias1 = b2[f * DTOK + nbase + 16 + l16];
    const float bias2 = b2[f * DTOK + nbase + 32 + l16];

    const int mrow0 = b0 + msub + half * 8;          // M-tile 0 base row for this lane
    #pragma unroll
    for (int r = 0; r < 8; ++r) {
        size_t base0 = (size_t)(mrow0 + r) * (NFEAT * DTOK) + (size_t)f * DTOK + nbase + l16;
        __builtin_nontemporal_store(acc00[r] + bias0, out + base0 +  0);
        __builtin_nontemporal_store(acc01[r] + bias1, out + base0 + 16);
        __builtin_nontemporal_store(acc02[r] + bias2, out + base0 + 32);
        size_t base1 = base0 + (size_t)16 * (NFEAT * DTOK);   // M-tile 1 (rows +16)
        __builtin_nontemporal_store(acc10[r] + bias0, out + base1 +  0);
        __builtin_nontemporal_store(acc11[r] + bias1, out + base1 + 16);
        __builtin_nontemporal_store(acc12[r] + bias2, out + base1 + 32);
    }
}

extern "C" void kernel_launch(void* const* d_in, const int* in_sizes, int n_in,
                              void* d_out, int out_size, void* d_ws, size_t ws_size,
                              hipStream_t stream) {
    const float* x  = (const float*)d_in[0];
    const float* W1 = (const float*)d_in[1];
    const float* b1 = (const float*)d_in[2];
    const float* W2 = (const float*)d_in[3];
    const float* b2 = (const float*)d_in[4];
    float* out = (float*)d_out;

    dim3 grid(BATCH / MTILE, NFEAT);   // (64, 100)
    dim3 block(256);
    mlp_tokenizer_wmma<<<grid, block, 0, stream>>>(x, W1, b1, W2, b2, out);
}